// TaggingFNNDecoder_CRF_6141803233743
// MI455X (gfx1250) — compile-verified
//
#include <hip/hip_runtime.h>
#include <hip/hip_bf16.h>
#include <stdint.h>

#define B_ 128
#define S_ 256
#define H_ 1024
#define T_ 64
#define TILE_K 128
#define PADK   (TILE_K + 4)     // 132-float LDS row stride: conflict-free b64 reads

typedef __attribute__((ext_vector_type(2))) float v2f;
typedef __attribute__((ext_vector_type(8))) float v8f;
typedef __attribute__((ext_vector_type(4))) int   v4i;

#ifndef __has_builtin
#define __has_builtin(x) 0
#endif

#if __has_builtin(__builtin_amdgcn_global_load_async_to_lds_b128)
#define HAVE_ASYNC_LDS 1
#else
#define HAVE_ASYNC_LDS 0
#endif

typedef __attribute__((address_space(1))) v4i* gv4i_p;   // global (AS1) v4i*
typedef __attribute__((address_space(3))) v4i* lv4i_p;   // LDS    (AS3) v4i*

__device__ __forceinline__ void wait_asynccnt0() {
#if __has_builtin(__builtin_amdgcn_s_wait_asynccnt)
    __builtin_amdgcn_s_wait_asynccnt(0);
#else
    asm volatile("s_wait_asynccnt 0x0" ::: "memory");
#endif
}

// ---------------------------------------------------------------------------
// Kernel 0: Wt[n][k] = W[k][n]  (one-time 256 KB transpose so GEMM staging is
// a straight async copy and B fragments are contiguous in K)
// ---------------------------------------------------------------------------
__global__ __launch_bounds__(256)
void transpose_W_kernel(const float* __restrict__ W, float* __restrict__ Wt) {
    const int idx = blockIdx.x * 256 + threadIdx.x;   // over T_*H_
    const int n = idx / H_;
    const int k = idx % H_;
    Wt[idx] = W[(size_t)k * T_ + n];
}

// ---------------------------------------------------------------------------
// Kernel 1: logits = hiddens @ W + b via V_WMMA_F32_16X16X4_F32.
// M = B*S = 32768, K = H = 1024, N = T = 64.
// 128 threads = 4 waves per block; each wave owns a 16(M) x 64(N) strip.
// Wt chunks (64 x 128 f32) staged to padded LDS via async-to-LDS (ASYNCcnt).
// ---------------------------------------------------------------------------
__global__ __launch_bounds__(128)
void gemm_logits_kernel(const float* __restrict__ hiddens,
                        const float* __restrict__ Wt,
                        const float* __restrict__ bias,
                        float* __restrict__ logits) {
    __shared__ float sWt[T_ * PADK];    // ~33 KB: sWt[n*PADK + k]

    const int lane = threadIdx.x & 31;
    const int wave = threadIdx.x >> 5;
    const int l15  = lane & 15;
    const int koff = (lane >> 4) << 1;              // lanes 0-15: K={0,1}; 16-31: K={2,3}
    const int m0   = (blockIdx.x * 4 + wave) * 16;

    v8f acc[4];
#pragma unroll
    for (int nt = 0; nt < 4; ++nt) acc[nt] = (v8f){0.f,0.f,0.f,0.f,0.f,0.f,0.f,0.f};

    const float* arow0 = hiddens + (size_t)(m0 + l15) * H_ + koff;

    for (int k0 = 0; k0 < H_; k0 += TILE_K) {
        __syncthreads();
        // stage Wt[0:64, k0:k0+128] -> sWt (padded rows), 16B per lane per iter
        for (int i = threadIdx.x; i < T_ * (TILE_K / 4); i += 128) {
            const int n  = i / (TILE_K / 4);
            const int kq = (i % (TILE_K / 4)) * 4;
            const float* g = Wt + (size_t)n * H_ + k0 + kq;
            float*       l = sWt + n * PADK + kq;
#if HAVE_ASYNC_LDS
            __builtin_amdgcn_global_load_async_to_lds_b128(
                (gv4i_p)(uintptr_t)g,
                (lv4i_p)(uint32_t)(uintptr_t)l, 0, 0);
#else
            *(float4*)l = *(const float4*)g;
#endif
        }
#if HAVE_ASYNC_LDS
        wait_asynccnt0();
#endif
        __syncthreads();

        const float* arow = arow0 + k0;
        __builtin_prefetch(arow + TILE_K, 0, 0);    // global_prefetch_b8: next A chunk

#pragma unroll 4
        for (int kk = 0; kk < TILE_K; kk += 4) {
            // A fragment: 16x4 f32 ISA layout, one b64 per lane
            v2f a = *(const v2f*)(arow + kk);
            // B fragments: single aligned ds_load_b64 each (padded stride 132)
            const float* bbase = sWt + kk + koff;
            v2f b0 = *(const v2f*)(bbase + (0 * 16 + l15) * PADK);
            v2f b1 = *(const v2f*)(bbase + (1 * 16 + l15) * PADK);
            v2f b2 = *(const v2f*)(bbase + (2 * 16 + l15) * PADK);
            v2f b3 = *(const v2f*)(bbase + (3 * 16 + l15) * PADK);
            acc[0] = __builtin_amdgcn_wmma_f32_16x16x4_f32(false, a, false, b0, (short)0, acc[0], false, false);
            acc[1] = __builtin_amdgcn_wmma_f32_16x16x4_f32(false, a, false, b1, (short)0, acc[1], false, false);
            acc[2] = __builtin_amdgcn_wmma_f32_16x16x4_f32(false, a, false, b2, (short)0, acc[2], false, false);
            acc[3] = __builtin_amdgcn_wmma_f32_16x16x4_f32(false, a, false, b3, (short)0, acc[3], false, false);
        }
    }

    // C/D layout: VGPR r holds (M = rbase + r, N = nt*16 + l15)
    const int rbase = m0 + ((lane >> 4) << 3);
#pragma unroll
    for (int nt = 0; nt < 4; ++nt) {
        const int col = nt * 16 + l15;
        const float bv = bias[col];
        float* outp = logits + (size_t)rbase * T_ + col;
#pragma unroll
        for (int r = 0; r < 8; ++r) outp[(size_t)r * T_] = acc[nt][r] + bv;
    }
}

// ---------------------------------------------------------------------------
// Kernel 2: fused forward-logsumexp + Viterbi scan. One 64-thread block per
// batch element; thread j owns tag j. Scores shared each step through LDS
// (broadcast reads). Thread 0 does the serial backtrace at the end.
// ---------------------------------------------------------------------------
__global__ __launch_bounds__(64)
void crf_scan_kernel(const float* __restrict__ logits,
                     const int*   __restrict__ mask,
                     const float* __restrict__ start,
                     const float* __restrict__ endt,
                     const float* __restrict__ trans,
                     uint8_t*     __restrict__ hist,
                     float*       __restrict__ den,
                     float*       __restrict__ pred) {
    __shared__ float strans[T_ * T_];   // 16 KB
    __shared__ float sf_s[T_];
    __shared__ float sv_s[T_];

    const int b = blockIdx.x;
    const int j = threadIdx.x;

    for (int i = j; i < T_ * T_; i += 64) strans[i] = trans[i];

    const float* lg = logits + (size_t)b * S_ * T_;
    const int*   mk = mask + (size_t)b * S_;

    float sf = start[j] + lg[j];
    float sv = sf;
    sf_s[j] = sf;
    sv_s[j] = sv;
    __syncthreads();

    for (int s = 1; s < S_; ++s) {
        const float e = lg[(size_t)s * T_ + j];
        const int   m = mk[s];

        // pass 1: max for logsumexp + max/argmax for Viterbi
        float mx = -3.0e38f, vmax = -3.0e38f;
        int varg = 0;
#pragma unroll 4
        for (int i = 0; i < T_; ++i) {
            const float t  = strans[i * T_ + j];
            const float vf = sf_s[i] + t;
            mx = fmaxf(mx, vf);
            const float vv = sv_s[i] + t;
            if (vv > vmax) { vmax = vv; varg = i; }
        }
        // pass 2: exp-sum
        float ssum = 0.0f;
#pragma unroll 4
        for (int i = 0; i < T_; ++i)
            ssum += __expf(sf_s[i] + strans[i * T_ + j] - mx);

        const float nf = mx + __logf(ssum) + e;
        const float nv = vmax + e;

        hist[((size_t)b * S_ + s) * T_ + j] = (uint8_t)varg;

        __syncthreads();                 // all lanes done reading old scores
        if (m) { sf = nf; sv = nv; }     // masked step keeps previous score
        sf_s[j] = sf;
        sv_s[j] = sv;
        __syncthreads();
    }

    sf_s[j] = sf + endt[j];
    sv_s[j] = sv + endt[j];
    __syncthreads();

    if (j == 0) {
        // denominator = logsumexp over tags
        float mx = sf_s[0];
        for (int i = 1; i < T_; ++i) mx = fmaxf(mx, sf_s[i]);
        float ssum = 0.0f;
        for (int i = 0; i < T_; ++i) ssum += __expf(sf_s[i] - mx);
        den[b] = mx + __logf(ssum);

        // Viterbi backtrace
        int cur = 0;
        float best = sv_s[0];
        for (int i = 1; i < T_; ++i)
            if (sv_s[i] > best) { best = sv_s[i]; cur = i; }
        for (int s = S_ - 1; s >= 1; --s) {
            const int m = mk[s];
            pred[(size_t)b * S_ + s] = (float)(m ? cur : 0);
            const int prev = hist[((size_t)b * S_ + s) * T_ + cur];
            if (m) cur = prev;
        }
        pred[(size_t)b * S_] = (float)cur;
    }
}

// ---------------------------------------------------------------------------
// Kernel 3: per-batch gold-path score (numerator) + loss reduction.
// ---------------------------------------------------------------------------
__global__ __launch_bounds__(B_)
void crf_num_loss_kernel(const float* __restrict__ logits,
                         const int*   __restrict__ mask,
                         const int*   __restrict__ labels,
                         const float* __restrict__ start,
                         const float* __restrict__ endt,
                         const float* __restrict__ trans,
                         const float* __restrict__ den,
                         float*       __restrict__ loss_out) {
    __shared__ float red[B_];
    const int b = threadIdx.x;

    const float* lg  = logits + (size_t)b * S_ * T_;
    const int*   lab = labels + (size_t)b * S_;
    const int*   mk  = mask + (size_t)b * S_;

    int   prev  = lab[0];
    float score = start[prev] + lg[prev];
    int   cnt   = mk[0];
    for (int s = 1; s < S_; ++s) {
        const int m   = mk[s];
        const int cur = lab[s];
        cnt += m;
        if (m) score += trans[prev * T_ + cur] + lg[(size_t)s * T_ + cur];
        prev = cur;
    }
    score += endt[lab[cnt - 1]];

    red[b] = score - den[b];
    __syncthreads();
    if (b == 0) {
        float tot = 0.0f;
        for (int i = 0; i < B_; ++i) tot += red[i];
        loss_out[0] = -tot;
    }
}

// ---------------------------------------------------------------------------
extern "C" void kernel_launch(void* const* d_in, const int* in_sizes, int n_in,
                              void* d_out, int out_size, void* d_ws, size_t ws_size,
                              hipStream_t stream) {
    (void)in_sizes; (void)n_in; (void)out_size; (void)ws_size;

    const float* hiddens = (const float*)d_in[0];   // (B,S,H) f32
    const int*   mask    = (const int*)  d_in[1];   // (B,S)  i32
    const int*   labels  = (const int*)  d_in[2];   // (B,S)  i32
    const float* W       = (const float*)d_in[3];   // (H,T)  f32
    const float* bias    = (const float*)d_in[4];   // (T,)   f32
    const float* start   = (const float*)d_in[5];   // (T,)   f32
    const float* endt    = (const float*)d_in[6];   // (T,)   f32
    const float* trans   = (const float*)d_in[7];   // (T,T)  f32

    float* out = (float*)d_out;                     // pred (B*S floats) ++ loss (1)

    // workspace: logits (8 MB f32) | hist (2 MB u8) | den (B f32) | Wt (256 KB)
    float*   logits = (float*)d_ws;
    uint8_t* hist   = (uint8_t*)(logits + (size_t)B_ * S_ * T_);
    float*   den    = (float*)(hist + (size_t)B_ * S_ * T_);
    float*   Wt     = den + B_;

    transpose_W_kernel<<<(H_ * T_) / 256, 256, 0, stream>>>(W, Wt);
    gemm_logits_kernel<<<(B_ * S_) / 64, 128, 0, stream>>>(hiddens, Wt, bias, logits);
    crf_scan_kernel<<<B_, 64, 0, stream>>>(logits, mask, start, endt, trans, hist, den, out);
    crf_num_loss_kernel<<<1, B_, 0, stream>>>(logits, mask, labels, start, endt, trans, den,
                                              out + (size_t)B_ * S_);
}